// MMDLoss_32856499814901
// MI455X (gfx1250) — compile-verified
//
#include <hip/hip_runtime.h>
#include <hip/hip_bf16.h>

// ---------------------------------------------------------------------------
// MMD loss (MI455X / gfx1250, wave32, WMMA f32 16x16x4 + async LDS copies)
//
// Passes (all on `stream`):
//   0. memset ws control region
//   1. sqnorm:  sq[i] = ||total_i||^2
//   2. gemm<0>: G tiles -> d2 -> histogram of top 12 float bits  (4096 bins)
//   3. scan(0): select bin b1, residual rank
//   4. gemm<1>: d2 -> histogram of bits[19:8] where top12==b1    (4096 bins)
//   5. scan(1): select b2, residual rank
//   6. gemm<2>: d2 -> histogram of bits[7:0] where bits[31:8]==b1b2 (256 bins)
//   7. scan(2): exact median bits -> bandwidth -> 5 inverse bandwidths
//   8. gemm<3>: d2 -> sum_q exp(-d2*ibw[q]) -> quadrant sums (f64 atomics)
//   9. final:   mmd = exx + eyy - (exy+eyx) ; clamp >= 0 ; write f32
// ---------------------------------------------------------------------------

#define NS_   4096
#define NT_   4096
#define NTOT  8192
#define DIM   512
#define BLK   128
#define KCH   16
#define KP    20          // padded LDS row stride (floats): 80B rows, 16B aligned
#define KRANK 33558527u   // n + (n*n - n - 1)/2 for n = 8192

// workspace layout (u32 units)
#define OFF_H0   0        // 4096 u32
#define OFF_H1   4096     // 4096 u32
#define OFF_H2   8192     // 4096 u32 (256 used)
#define OFF_CTRL 12288    // 32 u32: [0]=b1 [1]=rank1 [2]=b2 [3]=rank2, [8..12]=inv_bw f32
#define OFF_ACC  12320    // 3 doubles (byte 49280, 8B aligned)
#define OFF_SQN  12336    // 8192 f32
#define ZERO_BYTES (OFF_SQN * 4)

typedef float v2f __attribute__((ext_vector_type(2)));
typedef float v8f __attribute__((ext_vector_type(8)));
typedef int   v4i __attribute__((ext_vector_type(4)));

typedef __attribute__((address_space(1))) v4i  as1_v4i;
typedef __attribute__((address_space(3))) v4i  as3_v4i;
typedef __attribute__((address_space(3))) void as3_void;

// -------------------------------------------- async global->LDS B128 copy --
__device__ __forceinline__ void async_copy_b128(const float* g, float* l) {
#if __has_builtin(__builtin_amdgcn_global_load_async_to_lds_b128)
  __builtin_amdgcn_global_load_async_to_lds_b128((as1_v4i*)g, (as3_v4i*)l, 0, 0);
#else
  unsigned loff = (unsigned)(__UINTPTR_TYPE__)(as3_void*)l;
  asm volatile("global_load_async_to_lds_b128 %0, %1, off"
               :: "v"(loff), "v"(g) : "memory");
#endif
}

__device__ __forceinline__ void wait_async0() {
#if __has_builtin(__builtin_amdgcn_s_wait_asynccnt)
  __builtin_amdgcn_s_wait_asynccnt(0);
#else
  asm volatile("s_wait_asynccnt 0x0" ::: "memory");
#endif
}

// -------------------------------------------------------------- sqnorms ----
__global__ __launch_bounds__(256)
void sqnorm_kernel(const float* __restrict__ src, const float* __restrict__ tgt,
                   float* __restrict__ sqn) {
  int row  = blockIdx.x * 8 + (threadIdx.x >> 5);
  int lane = threadIdx.x & 31;
  const float* p = (row < NS_) ? (src + (size_t)row * DIM)
                               : (tgt + (size_t)(row - NS_) * DIM);
  float s = 0.f;
  for (int k = lane * 4; k < DIM; k += 32 * 4) {
    float4 v = *(const float4*)(p + k);
    s += v.x * v.x + v.y * v.y + v.z * v.z + v.w * v.w;
  }
  for (int off = 16; off > 0; off >>= 1) s += __shfl_down(s, off, 32);
  if (lane == 0) sqn[row] = s;
}

// ------------------------------------------------------------ main GEMM ----
// MODE 0/1/2: histogram passes; MODE 3: exp-kernel accumulation pass.
template <int MODE>
__global__ __launch_bounds__(256)
void gemm_kernel(const float* __restrict__ src, const float* __restrict__ tgt,
                 const float* __restrict__ sqn, unsigned* __restrict__ ghist,
                 const unsigned* __restrict__ ctrl, double* __restrict__ gacc) {
  // two double-buffered (A,B) chunk pairs: 2 * 2 * 128 * 20 * 4 = 40 KB
  __shared__ __align__(16) float smem[2 * 2 * BLK * KP];

  const int t    = threadIdx.x;
  const int lane = t & 31;
  const int w    = t >> 5;
  const int wr   = w >> 2;      // 0..1 : 64-row group
  const int wc   = w & 3;       // 0..3 : 32-col group
  const int half = lane >> 4;   // 0..1
  const int lm   = lane & 15;

  const int i0 = blockIdx.y * BLK;
  const int j0 = blockIdx.x * BLK;
  // 128-row tiles never straddle the source/target boundary (128 | 4096)
  const float* Abase = (i0 < NS_) ? (src + (size_t)i0 * DIM)
                                  : (tgt + (size_t)(i0 - NS_) * DIM);
  const float* Bbase = (j0 < NS_) ? (src + (size_t)j0 * DIM)
                                  : (tgt + (size_t)(j0 - NS_) * DIM);

  // stage one K-chunk (A and B tiles) into LDS buffer `buf` via async DMA
  auto stage = [&](int buf, int kc) {
#pragma unroll
    for (int rep = 0; rep < 2; ++rep) {
      int lin = t + rep * 256;           // 0..511
      int r   = lin >> 2;                // 0..127
      int c   = (lin & 3) * 4;           // 0,4,8,12  (16B aligned everywhere)
      float* base = smem + buf * (2 * BLK * KP);
      async_copy_b128(Abase + (size_t)r * DIM + kc + c, base + r * KP + c);
      async_copy_b128(Bbase + (size_t)r * DIM + kc + c, base + BLK * KP + r * KP + c);
    }
  };

  v8f acc[4][2] = {};

  stage(0, 0);
  int ib = 0;
  for (int kc = 0; kc < DIM; kc += KCH, ib ^= 1) {
    wait_async0();        // my async copies into buffer `ib` have landed
    __syncthreads();      // everyone's have; everyone done computing on ib^1
    if (kc + KCH < DIM) stage(ib ^ 1, kc + KCH);   // prefetch next chunk

    const float* As = smem + ib * (2 * BLK * KP);
    const float* Bs = As + BLK * KP;
#pragma unroll
    for (int ks = 0; ks < KCH; ks += 4) {
      const int kb = ks + 2 * half;      // lane-half selects K pair {0,1}/{2,3}
      v2f a[4], b[2];
#pragma unroll
      for (int mi = 0; mi < 4; ++mi)
        a[mi] = *(const v2f*)&As[(wr * 64 + mi * 16 + lm) * KP + kb];
#pragma unroll
      for (int ni = 0; ni < 2; ++ni)
        b[ni] = *(const v2f*)&Bs[(wc * 32 + ni * 16 + lm) * KP + kb];
#pragma unroll
      for (int mi = 0; mi < 4; ++mi)
#pragma unroll
        for (int ni = 0; ni < 2; ++ni)
          acc[mi][ni] = __builtin_amdgcn_wmma_f32_16x16x4_f32(
              false, a[mi], false, b[ni], (short)0, acc[mi][ni], false, false);
    }
  }
  __syncthreads();   // GEMM done; LDS free for epilogue reuse

  if constexpr (MODE <= 2) {
    const int NB = (MODE == 2) ? 256 : 4096;
    unsigned* hist = (unsigned*)smem;
    for (int b = t; b < NB; b += 256) hist[b] = 0;
    unsigned b1 = 0, b12 = 0;
    if (MODE == 1) b1  = ctrl[0];
    if (MODE == 2) b12 = (ctrl[0] << 12) | ctrl[2];
    __syncthreads();
#pragma unroll
    for (int mi = 0; mi < 4; ++mi)
#pragma unroll
      for (int ni = 0; ni < 2; ++ni)
#pragma unroll
        for (int r = 0; r < 8; ++r) {
          int gi = i0 + wr * 64 + mi * 16 + half * 8 + r;
          int gj = j0 + wc * 32 + ni * 16 + lm;
          float d2 = sqn[gi] + sqn[gj] - 2.0f * acc[mi][ni][r];
          d2 = fmaxf(d2, 0.0f);
          if (gi == gj) d2 = 0.0f;
          unsigned bits = __float_as_uint(d2);
          if (MODE == 0) {
            atomicAdd(&hist[bits >> 20], 1u);
          } else if (MODE == 1) {
            if ((bits >> 20) == b1) atomicAdd(&hist[(bits >> 8) & 0xFFFu], 1u);
          } else {
            if ((bits >> 8) == b12) atomicAdd(&hist[bits & 0xFFu], 1u);
          }
        }
    __syncthreads();
    for (int b = t; b < NB; b += 256) {
      unsigned c = hist[b];
      if (c) atomicAdd(&ghist[b], c);
    }
  } else {
    float ibw[5];
    const float* ibp = (const float*)(ctrl + 8);
#pragma unroll
    for (int q = 0; q < 5; ++q) ibw[q] = ibp[q];
    float sxx = 0.f, syy = 0.f, sxy = 0.f;
#pragma unroll
    for (int mi = 0; mi < 4; ++mi)
#pragma unroll
      for (int ni = 0; ni < 2; ++ni)
#pragma unroll
        for (int r = 0; r < 8; ++r) {
          int gi = i0 + wr * 64 + mi * 16 + half * 8 + r;
          int gj = j0 + wc * 32 + ni * 16 + lm;
          float d2 = sqn[gi] + sqn[gj] - 2.0f * acc[mi][ni][r];
          d2 = fmaxf(d2, 0.0f);
          if (gi == gj) d2 = 0.0f;
          float s = 0.f;
#pragma unroll
          for (int q = 0; q < 5; ++q) s += __expf(-d2 * ibw[q]);
          if (gi < NS_) { if (gj < NS_) sxx += s; else sxy += s; }
          else          { if (gj < NS_) sxy += s; else syy += s; }
        }
    double* red = (double*)smem;
    double v3[3] = {(double)sxx, (double)syy, (double)sxy};
    for (int q = 0; q < 3; ++q) {
      __syncthreads();
      red[t] = v3[q];
      __syncthreads();
      for (int off = 128; off > 0; off >>= 1) {
        if (t < off) red[t] += red[t + off];
        __syncthreads();
      }
      if (t == 0) atomicAdd(&gacc[q], red[0]);
    }
  }
}

// ------------------------------------------------------- radix-select scan -
__global__ void scan_kernel(unsigned* __restrict__ ws, int stage) {
  const unsigned* h = ws + (stage == 0 ? OFF_H0 : stage == 1 ? OFF_H1 : OFF_H2);
  unsigned* ctrl = ws + OFF_CTRL;
  int nb = (stage == 2) ? 256 : 4096;
  unsigned long long rank =
      (stage == 0) ? (unsigned long long)KRANK : (unsigned long long)ctrl[2 * stage - 1];
  unsigned long long cum = 0;
  int sel = 0;
  unsigned rem = 0;
  for (int b = 0; b < nb; ++b) {
    unsigned c = h[b];
    if (rank < cum + c) { sel = b; rem = (unsigned)(rank - cum); break; }
    cum += c;
  }
  if (stage < 2) {
    ctrl[2 * stage]     = (unsigned)sel;
    ctrl[2 * stage + 1] = rem;
  } else {
    unsigned bits = (ctrl[0] << 20) | (ctrl[2] << 8) | (unsigned)sel;
    float med = __uint_as_float(bits);
    float bw = (med + 1e-6f) * 0.25f;   // / KERNEL_MUL^(KERNEL_NUM//2) = /4
    float* f = (float*)(ctrl + 8);
    float m = 1.0f;
    for (int q = 0; q < 5; ++q) { f[q] = 1.0f / (bw * m); m *= 2.0f; }
  }
}

// ------------------------------------------------------------- finalize ----
__global__ void final_kernel(const double* __restrict__ gacc, float* __restrict__ out) {
  double exx = gacc[0] / ((double)NS_ * (double)NS_);
  double eyy = gacc[1] / ((double)NT_ * (double)NT_);
  double ecr = gacc[2] / ((double)NS_ * (double)NT_);  // e_xy + e_yx combined
  double mmd = exx + eyy - ecr;
  out[0] = (float)(mmd > 0.0 ? mmd : 0.0);
}

// ----------------------------------------------------------------- host ----
extern "C" void kernel_launch(void* const* d_in, const int* in_sizes, int n_in,
                              void* d_out, int out_size, void* d_ws, size_t ws_size,
                              hipStream_t stream) {
  const float* src = (const float*)d_in[0];
  const float* tgt = (const float*)d_in[1];
  float* out = (float*)d_out;

  unsigned* ws    = (unsigned*)d_ws;
  unsigned* hist0 = ws + OFF_H0;
  unsigned* hist1 = ws + OFF_H1;
  unsigned* hist2 = ws + OFF_H2;
  unsigned* ctrl  = ws + OFF_CTRL;
  double*   gacc  = (double*)(ws + OFF_ACC);
  float*    sqn   = (float*)(ws + OFF_SQN);

  (void)hipMemsetAsync(d_ws, 0, ZERO_BYTES, stream);

  sqnorm_kernel<<<NTOT / 8, 256, 0, stream>>>(src, tgt, sqn);

  dim3 grid(NTOT / BLK, NTOT / BLK);
  gemm_kernel<0><<<grid, 256, 0, stream>>>(src, tgt, sqn, hist0, ctrl, gacc);
  scan_kernel<<<1, 1, 0, stream>>>(ws, 0);
  gemm_kernel<1><<<grid, 256, 0, stream>>>(src, tgt, sqn, hist1, ctrl, gacc);
  scan_kernel<<<1, 1, 0, stream>>>(ws, 1);
  gemm_kernel<2><<<grid, 256, 0, stream>>>(src, tgt, sqn, hist2, ctrl, gacc);
  scan_kernel<<<1, 1, 0, stream>>>(ws, 2);
  gemm_kernel<3><<<grid, 256, 0, stream>>>(src, tgt, sqn, hist0, ctrl, gacc);
  final_kernel<<<1, 1, 0, stream>>>(gacc, out);
}